// PNANet_86852828659819
// MI455X (gfx1250) — compile-verified
//
#include <hip/hip_runtime.h>
#include <math.h>

#define NNODES 50000
#define NEDGES 120000
#define NGRAPH 2000
#define HID    128
#define TOW    4
#define KT     64
#define LDSROW 68   /* even pitch: b64-aligned LDS reads, conflict-free banks */
#define AVG_DEG_LOG 1.2242211692160564f

typedef __attribute__((ext_vector_type(2))) float v2f;
typedef __attribute__((ext_vector_type(8))) float v8f;

// ---------- order-preserving float<->uint encoding for atomic min/max ----------
__device__ __forceinline__ unsigned encOrd(float f) {
  unsigned u = __float_as_uint(f);
  return (u & 0x80000000u) ? ~u : (u | 0x80000000u);
}
__device__ __forceinline__ float decOrd(unsigned u) {
  return (u & 0x80000000u) ? __uint_as_float(u & 0x7FFFFFFFu) : __uint_as_float(~u);
}
#define ENC_PINF 0xFF800000u   /* encOrd(+inf): init for running-min */
#define ENC_NINF 0x007FFFFFu   /* encOrd(-inf): init for running-max */

// ---------- one-hot feature embedding (atoms: 7 feats -> 40 dims) ----------
__global__ void k_embed_atoms(const int* __restrict__ x,
                              const float* t0, const float* t1, const float* t2,
                              const float* t3, const float* t4, const float* t5,
                              const float* t6,
                              float* __restrict__ h, int N) {
  int n = blockIdx.x * blockDim.x + threadIdx.x;
  if (n >= N) return;
  const float* tabs[7] = {t0, t1, t2, t3, t4, t5, t6};
  const int dims[7] = {11, 7, 5, 6, 2, 2, 7};
  float buf[40];
  int off = 0; float tot = 0.f;
  for (int i = 0; i < 7; ++i) {
    int idx = x[n * 7 + i];
    const float* row = tabs[i] + idx * dims[i];
    float nr = 0.f;
    for (int j = 0; j < dims[i]; ++j) nr += row[j] * row[j];
    nr = sqrtf(nr);
    float sc = fminf(1.0f, 1.0f / fmaxf(nr, 1e-7f));
    for (int j = 0; j < dims[i]; ++j) { float v = row[j] * sc; buf[off + j] = v; tot += v * v; }
    off += dims[i];
  }
  float inv = 1.0f / fmaxf(sqrtf(tot), 1e-12f);
  for (int j = 0; j < 40; ++j) h[(size_t)n * 40 + j] = buf[j] * inv;
}

// ---------- edge embedding (4 feats -> 10 dims) ----------
__global__ void k_embed_edges(const int* __restrict__ x,
                              const float* t0, const float* t1, const float* t2,
                              const float* t3,
                              float* __restrict__ e, int E) {
  int n = blockIdx.x * blockDim.x + threadIdx.x;
  if (n >= E) return;
  const float* tabs[4] = {t0, t1, t2, t3};
  const int dims[4] = {4, 2, 2, 2};
  float buf[10];
  int off = 0; float tot = 0.f;
  for (int i = 0; i < 4; ++i) {
    int idx = x[n * 4 + i];
    const float* row = tabs[i] + idx * dims[i];
    float nr = 0.f;
    for (int j = 0; j < dims[i]; ++j) nr += row[j] * row[j];
    nr = sqrtf(nr);
    float sc = fminf(1.0f, 1.0f / fmaxf(nr, 1e-7f));
    for (int j = 0; j < dims[i]; ++j) { float v = row[j] * sc; buf[off + j] = v; tot += v * v; }
    off += dims[i];
  }
  float inv = 1.0f / fmaxf(sqrtf(tot), 1e-12f);
  for (int j = 0; j < 10; ++j) e[(size_t)n * 10 + j] = buf[j] * inv;
}

// ---------- pad + transpose weights: Bp[t][n][k] (Npad x Kpad, zero-filled) ----------
__global__ void k_padBT(const float* __restrict__ B, float* __restrict__ Bp,
                        int T, int K, int N, int Kpad, int Npad) {
  int tot = T * Npad * Kpad;
  int i = blockIdx.x * blockDim.x + threadIdx.x;
  if (i >= tot) return;
  int t = i / (Npad * Kpad);
  int r = i % (Npad * Kpad);
  int n = r / Kpad;
  int k = r % Kpad;
  Bp[i] = (k < K && n < N) ? B[((size_t)t * K + k) * N + n] : 0.f;
}

// ---------- generic WMMA f32 GEMM, one wave per 16x16 C tile ----------
// MODE 0: A = dense MxK (lda)
// MODE 1: A row = edge; cols = [h[col[e]] | h[row[e]] | eL[e]]  (per-tower slice of h)
// MODE 2: A row = node; cols = [xt | {s,mean,min,max,std} x {1, amp, 1/amp}]
// Bp is padded+transposed (t, Npad, Kpad); bias is the ORIGINAL (t, N) bias.
template <int MODE>
__global__ void k_gemm(const float* __restrict__ Bp, const float* __restrict__ bias,
                       float* __restrict__ out,
                       int M, int K, int N, int Kpad, int Npad, int ldo,
                       const float* __restrict__ A, int lda,
                       const float* __restrict__ h, int ldh,
                       const float* __restrict__ eL,
                       const int* __restrict__ eiSrc, const int* __restrict__ eiDst,
                       int ft,
                       const float* __restrict__ s, const float* __restrict__ s2,
                       const unsigned* __restrict__ mne, const unsigned* __restrict__ mxe,
                       const float* __restrict__ cnt) {
  __shared__ float Ats[16 * LDSROW];
  const int t    = blockIdx.z;
  const int m0   = blockIdx.x * 16;
  const int lane = threadIdx.x & 31;
  const int wave = threadIdx.x >> 5;
  const int col0 = wave * 16;
  const int koff = (lane >> 4) << 1;   // lanes 16-31 hold K+2 per ISA A/B layout
  const int row  = lane & 15;
  const int nn   = col0 + (lane & 15); // nn < Npad always
  const float* bb = Bp + ((size_t)t * Npad + nn) * Kpad;  // column nn of tower t
  v8f acc = {0.f, 0.f, 0.f, 0.f, 0.f, 0.f, 0.f, 0.f};

  for (int k0 = 0; k0 < Kpad; k0 += KT) {
    __syncthreads();
    // cooperative A-tile stage (gather / aggregate fused here); zero beyond M,K
    for (int idx = threadIdx.x; idx < 16 * KT; idx += blockDim.x) {
      int r = idx >> 6, kk = idx & (KT - 1);
      int gm = m0 + r, gk = k0 + kk;
      float v = 0.f;
      if (gm < M && gk < K) {
        if (MODE == 0) {
          v = A[(size_t)gm * lda + gk];
        } else if (MODE == 1) {
          if (gk < ft)            v = h[(size_t)eiDst[gm] * ldh + t * ft + gk];
          else if (gk < 2 * ft)   v = h[(size_t)eiSrc[gm] * ldh + t * ft + (gk - ft)];
          else                    v = eL[(size_t)gm * ft + (gk - 2 * ft)];
        } else {  // MODE 2
          int grp = gk / ft, c = gk % ft;
          if (grp == 0) {
            v = h[(size_t)gm * ldh + t * ft + c];
          } else {
            float cn    = cnt[gm];
            float denom = fmaxf(cn, 1.0f);
            size_t o    = ((size_t)gm * gridDim.z + t) * ft + c;
            int g2 = (grp - 1) % 5;
            float base;
            if (g2 == 0)      base = s[o];
            else if (g2 == 1) base = s[o] / denom;
            else if (g2 == 2) base = (cn > 0.f) ? decOrd(mne[o]) : 0.f;
            else if (g2 == 3) base = (cn > 0.f) ? decOrd(mxe[o]) : 0.f;
            else {
              float mean = s[o] / denom;
              float var  = s2[o] / denom - mean * mean;
              base = sqrtf(fmaxf(var, 0.f) + 1e-5f);
            }
            if (grp >= 11)     { float amp = logf(denom + 1.f) / AVG_DEG_LOG; v = base / amp; }
            else if (grp >= 6) { float amp = logf(denom + 1.f) / AVG_DEG_LOG; v = base * amp; }
            else               v = base;
          }
        }
      }
      Ats[r * LDSROW + kk] = v;
    }
    __syncthreads();

    const int kmax = (Kpad - k0 < KT) ? (Kpad - k0) : KT;  // multiple of 4
#pragma unroll 4
    for (int kk = 0; kk < kmax; kk += 4) {
      v2f a = *(const v2f*)&Ats[row * LDSROW + kk + koff];        // ds_load_b64
      v2f b = *(const v2f*)&bb[k0 + kk + koff];                   // global_load_b64
      acc = __builtin_amdgcn_wmma_f32_16x16x4_f32(false, a, false, b, (short)0, acc,
                                                  false, false);
    }
  }

  // epilogue: C/D layout = VGPR v -> row v (lanes 0-15) / row 8+v (lanes 16-31)
  int rbase = (lane >> 4) << 3;
  int c  = lane & 15;
  int gc = col0 + c;
  if (gc < N) {
    float bv = bias[(size_t)t * N + gc];
#pragma unroll
    for (int vg = 0; vg < 8; ++vg) {
      int gr = m0 + rbase + vg;
      if (gr < M) out[(size_t)gr * ldo + (size_t)t * N + gc] = acc[vg] + bv;
    }
  }
}

// ---------- aggregation buffers init ----------
__global__ void k_init_agg(float* s, float* s2, unsigned* mn, unsigned* mx, float* cnt,
                           size_t tot, int N) {
  size_t i = (size_t)blockIdx.x * blockDim.x + threadIdx.x;
  if (i < tot) { s[i] = 0.f; s2[i] = 0.f; mn[i] = ENC_PINF; mx[i] = ENC_NINF; }
  if (i < (size_t)N) cnt[i] = 0.f;
}

// ---------- edge -> node scatter (segment sum/sumsq/min/max/count) ----------
__global__ void k_scatter(const float* __restrict__ m, const int* __restrict__ dst,
                          int E, int TF,
                          float* s, float* s2, unsigned* mn, unsigned* mx, float* cnt) {
  size_t idx = (size_t)blockIdx.x * blockDim.x + threadIdx.x;
  size_t tot = (size_t)E * TF;
  if (idx >= tot) return;
  int e = (int)(idx / TF), j = (int)(idx % TF);
  float v = m[idx];
  int node = dst[e];
  size_t o = (size_t)node * TF + j;
  atomicAdd(&s[o], v);
  atomicAdd(&s2[o], v * v);
  unsigned ev = encOrd(v);
  atomicMin(&mn[o], ev);
  atomicMax(&mx[o], ev);
  if (j == 0) atomicAdd(&cnt[node], 1.0f);
}

// ---------- utility ----------
__global__ void k_zero(float* p, int n) {
  int i = blockIdx.x * blockDim.x + threadIdx.x;
  if (i < n) p[i] = 0.f;
}

// ---------- batch-norm statistics (blockDim.x == C) ----------
__global__ void k_bn_reduce(const float* __restrict__ h, int N, int C, float* stats) {
  int c = threadIdx.x;
  float sum = 0.f, sq = 0.f;
  for (int r = blockIdx.x; r < N; r += gridDim.x) {
    float v = h[(size_t)r * C + c];
    sum += v; sq += v * v;
  }
  atomicAdd(&stats[c], sum);
  atomicAdd(&stats[C + c], sq);
}

// ---------- BN + (leaky) ReLU; negslope=1 -> identity, 0 -> relu ----------
__global__ void k_bn_act(float* h, int N, int C, const float* __restrict__ stats,
                         const float* __restrict__ gamma, const float* __restrict__ beta,
                         float negslope) {
  size_t i = (size_t)blockIdx.x * blockDim.x + threadIdx.x;
  if (i >= (size_t)N * C) return;
  int c = (int)(i % C);
  float invN = 1.0f / (float)N;
  float mu  = stats[c] * invN;
  float var = stats[C + c] * invN - mu * mu;
  float y = gamma[c] * (h[i] - mu) * rsqrtf(var + 1e-5f) + beta[c];
  h[i] = (y >= 0.f) ? y : negslope * y;
}

// ---------- graph pooling ----------
__global__ void k_pool(const float* __restrict__ h, const int* __restrict__ bidx,
                       int N, int C, float* g) {
  size_t i = (size_t)blockIdx.x * blockDim.x + threadIdx.x;
  if (i >= (size_t)N * C) return;
  int n = (int)(i / C), c = (int)(i % C);
  atomicAdd(&g[(size_t)bidx[n] * C + c], h[i]);
}

// ---------- final 64 -> 2 head ----------
__global__ void k_final(const float* __restrict__ gin, const float* __restrict__ w3,
                        const float* __restrict__ b3, float* __restrict__ out, int G) {
  int i = blockIdx.x * blockDim.x + threadIdx.x;
  if (i >= G * 2) return;
  int g = i >> 1, o = i & 1;
  float acc = b3[o];
  for (int c = 0; c < 64; ++c) acc += gin[(size_t)g * 64 + c] * w3[c * 2 + o];
  out[i] = acc;
}

// ==========================================================================
extern "C" void kernel_launch(void* const* d_in, const int* in_sizes, int n_in,
                              void* d_out, int out_size, void* d_ws, size_t ws_size,
                              hipStream_t stream) {
  const int N = NNODES, E = NEDGES, G = NGRAPH;

  const int* x    = (const int*)d_in[0];
  const int* ei   = (const int*)d_in[1];   // (2, E): [0]=row(src), [1]=col(dst)
  const int* ea   = (const int*)d_in[2];
  const int* bidx = (const int*)d_in[3];
  const float* at[7]; for (int i = 0; i < 7; ++i) at[i] = (const float*)d_in[4 + i];
  const float* et[4]; for (int i = 0; i < 4; ++i) et[i] = (const float*)d_in[11 + i];
  // layer params: 15 + l*10 + {edge_w,edge_b,pre_w,pre_b,post_w,post_b,lin_w,lin_b,bn_g,bn_b}
  // mlp params: 55..64
  const float* w1  = (const float*)d_in[55]; const float* b1  = (const float*)d_in[56];
  const float* mg1 = (const float*)d_in[57]; const float* mb1 = (const float*)d_in[58];
  const float* w2  = (const float*)d_in[59]; const float* b2  = (const float*)d_in[60];
  const float* mg2 = (const float*)d_in[61]; const float* mb2 = (const float*)d_in[62];
  const float* w3  = (const float*)d_in[63]; const float* b3  = (const float*)d_in[64];
  float* outp = (float*)d_out;

  // ---- workspace carve-up ----
  char* w = (char*)d_ws;
  auto carve = [&](size_t bytes) { void* p = (void*)w; w += (bytes + 255) & ~(size_t)255; return p; };
  float*    hA    = (float*)carve((size_t)N * HID * 4);
  float*    hB    = (float*)carve((size_t)N * HID * 4);
  float*    eemb  = (float*)carve((size_t)E * 10 * 4);
  float*    eLbuf = (float*)carve((size_t)E * 32 * 4);
  float*    mbuf  = (float*)carve((size_t)E * HID * 4);   // messages; later reused as post-GEMM out
  float*    sbuf  = (float*)carve((size_t)N * HID * 4);
  float*    s2buf = (float*)carve((size_t)N * HID * 4);
  unsigned* mnbuf = (unsigned*)carve((size_t)N * HID * 4);
  unsigned* mxbuf = (unsigned*)carve((size_t)N * HID * 4);
  float*    cntb  = (float*)carve((size_t)N * 4);
  float*    stats = (float*)carve(256 * 4);
  float*    gpool = (float*)carve((size_t)G * HID * 4);
  float*    gtmp1 = (float*)carve((size_t)G * 64 * 4);
  float*    gtmp2 = (float*)carve((size_t)G * 64 * 4);
  float*    Bpbuf = (float*)carve((size_t)128 * 1024 * 4); // padded/transposed weights

  const int* eiSrc = ei;      // row
  const int* eiDst = ei + E;  // col (aggregation target)

  // ---- embeddings ----
  k_embed_atoms<<<(N + 255) / 256, 256, 0, stream>>>(x, at[0], at[1], at[2], at[3], at[4],
                                                     at[5], at[6], hA, N);
  k_embed_edges<<<(E + 255) / 256, 256, 0, stream>>>(ea, et[0], et[1], et[2], et[3], eemb, E);

  auto launch_gemm = [&](int mode, const float* B, const float* bias, float* out,
                         int M, int K, int Nc, int ldo, int towers,
                         const float* A, int lda,
                         const float* h, int ldh, const float* eL, int ft,
                         const float* s, const float* s2,
                         const unsigned* mn, const unsigned* mx, const float* cnt) {
    int Kpad = (K + 3) & ~3;
    int Npad = (Nc + 15) & ~15;
    int padTot = towers * Npad * Kpad;
    k_padBT<<<(padTot + 255) / 256, 256, 0, stream>>>(B, Bpbuf, towers, K, Nc, Kpad, Npad);
    dim3 grid((M + 15) / 16, 1, towers);
    dim3 blk(32 * (Npad / 16));
    if (mode == 0)
      k_gemm<0><<<grid, blk, 0, stream>>>(Bpbuf, bias, out, M, K, Nc, Kpad, Npad, ldo,
                                          A, lda, h, ldh, eL, eiSrc, eiDst, ft,
                                          s, s2, mn, mx, cnt);
    else if (mode == 1)
      k_gemm<1><<<grid, blk, 0, stream>>>(Bpbuf, bias, out, M, K, Nc, Kpad, Npad, ldo,
                                          A, lda, h, ldh, eL, eiSrc, eiDst, ft,
                                          s, s2, mn, mx, cnt);
    else
      k_gemm<2><<<grid, blk, 0, stream>>>(Bpbuf, bias, out, M, K, Nc, Kpad, Npad, ldo,
                                          A, lda, h, ldh, eL, eiSrc, eiDst, ft,
                                          s, s2, mn, mx, cnt);
  };

  const float* hin = hA;
  float* hout = hB;
  for (int l = 0; l < 4; ++l) {
    const int base = 15 + l * 10;
    const float* edge_w = (const float*)d_in[base + 0];
    const float* edge_b = (const float*)d_in[base + 1];
    const float* pre_w  = (const float*)d_in[base + 2];
    const float* pre_b  = (const float*)d_in[base + 3];
    const float* post_w = (const float*)d_in[base + 4];
    const float* post_b = (const float*)d_in[base + 5];
    const float* lin_w  = (const float*)d_in[base + 6];
    const float* lin_b  = (const float*)d_in[base + 7];
    const float* bn_g   = (const float*)d_in[base + 8];
    const float* bn_b   = (const float*)d_in[base + 9];

    const int fin = (l == 0) ? 40 : HID;
    const int ft  = fin / TOW;          // 10 or 32
    const int TF  = TOW * ft;

    // e_l = e_emb @ edge_w + edge_b            (E x 10) @ (10 x ft)
    launch_gemm(0, edge_w, edge_b, eLbuf, E, 10, ft, ft, 1,
                eemb, 10, nullptr, 0, nullptr, 0, nullptr, nullptr, nullptr, nullptr, nullptr);

    // m = [h[col]|h[row]|e_l] @ pre_w + pre_b  per tower (fused gather)
    launch_gemm(1, pre_w, pre_b, mbuf, E, 3 * ft, ft, TF, TOW,
                nullptr, 0, hin, fin, eLbuf, ft, nullptr, nullptr, nullptr, nullptr, nullptr);

    // segment aggregations
    size_t totA = (size_t)N * TF;
    k_init_agg<<<(unsigned)((totA + 255) / 256), 256, 0, stream>>>(sbuf, s2buf, mnbuf, mxbuf,
                                                                   cntb, totA, N);
    size_t totS = (size_t)E * TF;
    k_scatter<<<(unsigned)((totS + 255) / 256), 256, 0, stream>>>(mbuf, eiDst, E, TF, sbuf,
                                                                  s2buf, mnbuf, mxbuf, cntb);

    // post: [xt | 15 scaled aggregates] @ post_w + post_b  -> reuse mbuf as (N,128) out
    launch_gemm(2, post_w, post_b, mbuf, N, 16 * ft, 32, HID, TOW,
                nullptr, 0, hin, fin, nullptr, ft, sbuf, s2buf, mnbuf, mxbuf, cntb);

    // lin: (N,128) @ (128,128) + b
    launch_gemm(0, lin_w, lin_b, hout, N, HID, HID, HID, 1,
                mbuf, HID, nullptr, 0, nullptr, 0, nullptr, nullptr, nullptr, nullptr, nullptr);

    // batch-norm + leaky relu (alpha = 0.01)
    k_zero<<<1, 256, 0, stream>>>(stats, 2 * HID);
    k_bn_reduce<<<512, HID, 0, stream>>>(hout, N, HID, stats);
    size_t totH = (size_t)N * HID;
    k_bn_act<<<(unsigned)((totH + 255) / 256), 256, 0, stream>>>(hout, N, HID, stats, bn_g,
                                                                 bn_b, 0.01f);
    const float* tmp = hin; hin = hout; hout = (float*)tmp;
  }

  // ---- graph pooling ----
  k_zero<<<(G * HID + 255) / 256, 256, 0, stream>>>(gpool, G * HID);
  size_t totP = (size_t)N * HID;
  k_pool<<<(unsigned)((totP + 255) / 256), 256, 0, stream>>>(hin, bidx, N, HID, gpool);

  // ---- MLP head ----
  launch_gemm(0, w1, b1, gtmp1, G, HID, 64, 64, 1,
              gpool, HID, nullptr, 0, nullptr, 0, nullptr, nullptr, nullptr, nullptr, nullptr);
  k_zero<<<1, 256, 0, stream>>>(stats, 128);
  k_bn_reduce<<<512, 64, 0, stream>>>(gtmp1, G, 64, stats);
  k_bn_act<<<(G * 64 + 255) / 256, 256, 0, stream>>>(gtmp1, G, 64, stats, mg1, mb1, 0.0f);

  launch_gemm(0, w2, b2, gtmp2, G, 64, 64, 64, 1,
              gtmp1, 64, nullptr, 0, nullptr, 0, nullptr, nullptr, nullptr, nullptr, nullptr);
  k_zero<<<1, 256, 0, stream>>>(stats, 128);
  k_bn_reduce<<<512, 64, 0, stream>>>(gtmp2, G, 64, stats);
  k_bn_act<<<(G * 64 + 255) / 256, 256, 0, stream>>>(gtmp2, G, 64, stats, mg2, mb2, 0.0f);

  k_final<<<(G * 2 + 255) / 256, 256, 0, stream>>>(gtmp2, w3, b3, outp, G);
}